// Top2Router_41386304864538
// MI455X (gfx1250) — compile-verified
//
#include <hip/hip_runtime.h>

typedef __attribute__((ext_vector_type(2))) float v2f;
typedef __attribute__((ext_vector_type(8))) float v8f;

#define T_TOKENS 16384
#define D_DIM    4096
#define E_EXP    64
#define KC       64
#define LDS_STRIDE (KC + 4)     // 68 dwords: bank rotation of 4 per expert row -> conflict-free b64
#define NCHUNK   (D_DIM / KC)   // 64

// Async global -> LDS copy (CDNA5, tracked by ASYNCcnt). 16B per lane.
__device__ __forceinline__ void async_ld_b128(unsigned lds_off, const float* gptr) {
    unsigned long long addr = (unsigned long long)(uintptr_t)gptr;
    asm volatile("global_load_async_to_lds_b128 %0, %1, off"
                 :: "v"(lds_off), "v"(addr) : "memory");
}

__device__ __forceinline__ void wait_asynccnt0() {
    asm volatile("s_wait_asynccnt 0x0" ::: "memory");
}

// Stage W[:, kc:kc+KC] (64 x 64 f32) into LDS buffer at byte offset lds_base.
// 1024 float4 tiles / 256 threads = 4 per thread; 16 threads cover one row (256B) -> coalesced.
__device__ __forceinline__ void stage_chunk_async(const float* __restrict__ W, int kc,
                                                  unsigned lds_base, int tid) {
    #pragma unroll
    for (int i = 0; i < 4; ++i) {
        int idx = tid + i * 256;          // 0..1023
        int e   = idx >> 4;               // / (KC/4)
        int c4  = idx & 15;
        unsigned loff = lds_base + (unsigned)((e * LDS_STRIDE + c4 * 4) * 4);
        async_ld_b128(loff, W + (size_t)e * D_DIM + kc + c4 * 4);
    }
}

__global__ __launch_bounds__(256) void top2router_kernel(
    const float* __restrict__ x, const float* __restrict__ W,
    const float* __restrict__ b, float* __restrict__ out)
{
    __shared__ float sW[2][E_EXP * LDS_STRIDE];   // 2 x 17 KB, double-buffered

    const int tid    = threadIdx.x;
    const int lane   = tid & 31;
    const int wave   = tid >> 5;
    const int lane16 = lane & 15;
    const int hi     = lane >> 4;              // 0: lanes 0-15, 1: lanes 16-31
    const int t0     = blockIdx.x * 128 + wave * 16;

    // A fragment (16x4 fp32): lane<16 -> row t0+lane16, cols k..k+1 ; lane>=16 -> cols k+2..k+3
    const float* xrow = x + (size_t)(t0 + lane16) * D_DIM + 2 * hi;

    const unsigned lds0 = (unsigned)(uintptr_t)(&sW[0][0]);
    const unsigned lds1 = (unsigned)(uintptr_t)(&sW[1][0]);

    v8f acc0 = {}, acc1 = {}, acc2 = {}, acc3 = {};

    // Per-lane B-fragment offsets within a chunk buffer (dwords)
    const int o0 = ( 0 + lane16) * LDS_STRIDE + 2 * hi;
    const int o1 = (16 + lane16) * LDS_STRIDE + 2 * hi;
    const int o2 = (32 + lane16) * LDS_STRIDE + 2 * hi;
    const int o3 = (48 + lane16) * LDS_STRIDE + 2 * hi;

    // Prime the pipeline: chunk 0 -> buffer 0
    stage_chunk_async(W, 0, lds0, tid);

    for (int c = 0; c < NCHUNK; ++c) {
        wait_asynccnt0();          // my async copies for chunk c are in LDS
        __syncthreads();           // everyone's copies done; everyone done reading other buffer

        if (c + 1 < NCHUNK)        // overlap: stage chunk c+1 into the other buffer
            stage_chunk_async(W, (c + 1) * KC, ((c + 1) & 1) ? lds1 : lds0, tid);

        const float* sbuf = &sW[c & 1][0];
        const float* xp   = xrow + c * KC;

        #pragma unroll 4
        for (int kk = 0; kk < KC; kk += 4) {
            v2f a  = __builtin_nontemporal_load((const v2f*)(xp + kk));
            v2f b0 = *(const v2f*)(sbuf + o0 + kk);
            v2f b1 = *(const v2f*)(sbuf + o1 + kk);
            v2f b2 = *(const v2f*)(sbuf + o2 + kk);
            v2f b3 = *(const v2f*)(sbuf + o3 + kk);
            acc0 = __builtin_amdgcn_wmma_f32_16x16x4_f32(false, a, false, b0, (short)0, acc0, false, false);
            acc1 = __builtin_amdgcn_wmma_f32_16x16x4_f32(false, a, false, b1, (short)0, acc1, false, false);
            acc2 = __builtin_amdgcn_wmma_f32_16x16x4_f32(false, a, false, b2, (short)0, acc2, false, false);
            acc3 = __builtin_amdgcn_wmma_f32_16x16x4_f32(false, a, false, b3, (short)0, acc3, false, false);
        }
    }

    // Bias (zero in reference, but stay faithful)
    const float bias0 = b[ 0 + lane16];
    const float bias1 = b[16 + lane16];
    const float bias2 = b[32 + lane16];
    const float bias3 = b[48 + lane16];

    const int e0 = lane16, e1 = 16 + lane16, e2 = 32 + lane16, e3 = 48 + lane16;

    // Epilogue: each 16-lane half-row holds all 64 logits of one token (component r).
    #pragma unroll
    for (int r = 0; r < 8; ++r) {
        float v0 = acc0[r] + bias0;
        float v1 = acc1[r] + bias1;
        float v2 = acc2[r] + bias2;
        float v3 = acc3[r] + bias3;

        // local top-2 of 4
        float pav = v0 >= v1 ? v0 : v1;  int pai = v0 >= v1 ? e0 : e1;
        float pbv = v0 >= v1 ? v1 : v0;  int pbi = v0 >= v1 ? e1 : e0;
        float qav = v2 >= v3 ? v2 : v3;  int qai = v2 >= v3 ? e2 : e3;
        float qbv = v2 >= v3 ? v3 : v2;  int qbi = v2 >= v3 ? e3 : e2;
        float t1v, t2v; int t1i, t2i;
        if (pav >= qav) {
            t1v = pav; t1i = pai;
            t2v = pbv >= qav ? pbv : qav;  t2i = pbv >= qav ? pbi : qai;
        } else {
            t1v = qav; t1i = qai;
            t2v = qbv >= pav ? qbv : pav;  t2i = qbv >= pav ? qbi : pai;
        }

        // butterfly top-2 merge across the 16-lane group (masks 1..8 keep halves separate)
        #pragma unroll
        for (int m = 1; m <= 8; m <<= 1) {
            float o1v = __shfl_xor(t1v, m, 32);  int o1i = __shfl_xor(t1i, m, 32);
            float o2v = __shfl_xor(t2v, m, 32);  int o2i = __shfl_xor(t2i, m, 32);
            if (o1v > t1v) {
                float n2v = t1v >= o2v ? t1v : o2v;  int n2i = t1v >= o2v ? t1i : o2i;
                t1v = o1v; t1i = o1i; t2v = n2v; t2i = n2i;
            } else {
                if (o1v > t2v) { t2v = o1v; t2i = o1i; }
            }
        }

        // softmax denominator (global max == t1v)
        float mx = t1v;
        float s = __expf(v0 - mx) + __expf(v1 - mx) + __expf(v2 - mx) + __expf(v3 - mx);
        #pragma unroll
        for (int m = 1; m <= 8; m <<= 1) s += __shfl_xor(s, m, 32);

        float p1 = __expf(t1v - mx) / s;              // top-1 probability
        float p2 = __expf(t2v - mx) / s;              // top-2 probability
        float w1 = 1.0f / (1.0f + __expf(p2 - p1));   // softmax over the two probs
        float w2 = 1.0f - w1;

        const int tok = t0 + r + 8 * hi;
        float* orow = out + (size_t)tok * E_EXP;
        orow[e0] = (e0 == t1i) ? w1 : ((e0 == t2i) ? w2 : 0.0f);
        orow[e1] = (e1 == t1i) ? w1 : ((e1 == t2i) ? w2 : 0.0f);
        orow[e2] = (e2 == t1i) ? w1 : ((e2 == t2i) ? w2 : 0.0f);
        orow[e3] = (e3 == t1i) ? w1 : ((e3 == t2i) ? w2 : 0.0f);
    }
}

extern "C" void kernel_launch(void* const* d_in, const int* in_sizes, int n_in,
                              void* d_out, int out_size, void* d_ws, size_t ws_size,
                              hipStream_t stream) {
    const float* x = (const float*)d_in[0];
    const float* W = (const float*)d_in[1];
    const float* b = (const float*)d_in[2];
    float* out = (float*)d_out;
    (void)in_sizes; (void)n_in; (void)out_size; (void)d_ws; (void)ws_size;

    dim3 grid(T_TOKENS / 128);   // 128 tokens per block (8 waves x 16 tokens)
    dim3 block(256);
    top2router_kernel<<<grid, block, 0, stream>>>(x, W, b, out);
}